// ModelNew_66924180406691
// MI455X (gfx1250) — compile-verified
//
#include <hip/hip_runtime.h>

typedef __attribute__((ext_vector_type(2))) float v2f;
typedef __attribute__((ext_vector_type(8))) float v8f;

constexpr int BATCH = 8;
constexpr int SEQ   = 8192;
constexpr int NH    = 16;     // heads
constexpr int DP    = 64;     // d_head
constexpr int DN    = 64;     // d_state
constexpr int CL    = 64;     // chunk length
constexpr int NC    = SEQ / CL;   // 128 chunks
constexpr int CPB   = 8;          // chunks per block
constexpr int CG    = NC / CPB;   // 16 chunk-groups

// ---------------- kernel 0: zero the output (atomics accumulate into it) ----
__global__ void zero_out_kernel(float* __restrict__ out, int n) {
    int i = blockIdx.x * blockDim.x + threadIdx.x;
    if (i < n) out[i] = 0.0f;
}

// ---------------- kernel 1: per-(b,h) chunk-end sums + inclusive suffix scan
// Tsuf[bh*NC + c] = sum_{c' >= c} sum_l A[b, c'*CL + l, h]
__global__ __launch_bounds__(NC) void suffix_scan_kernel(const float* __restrict__ A,
                                                         float* __restrict__ Tsuf) {
    int bh = blockIdx.x;                 // b*NH + h
    int b  = bh / NH;
    int h  = bh % NH;
    int c  = threadIdx.x;                // chunk id, 0..NC-1

    __shared__ float ends[NC];
    const float* Ab = A + ((size_t)b * SEQ) * NH + h;
    float s = 0.0f;
    int t0 = c * CL;
    for (int l = 0; l < CL; ++l) s += Ab[(size_t)(t0 + l) * NH];
    ends[c] = s;
    __syncthreads();

    float suf = 0.0f;
    for (int cc = c; cc < NC; ++cc) suf += ends[cc];   // inclusive suffix sum
    Tsuf[(size_t)bh * NC + c] = suf;
}

// -------- helpers for the CDNA5 async global->LDS path (ASYNCcnt-tracked) --
__device__ __forceinline__ unsigned lds_addr32(const void* p) {
    // LDS aperture: hardware uses addr[31:0] as the DS byte address
    return (unsigned)(unsigned long long)(uintptr_t)p;
}

__device__ __forceinline__ void async_b128_to_lds(unsigned lds_byte, const void* gaddr) {
    asm volatile("global_load_async_to_lds_b128 %0, %1, off"
                 :: "v"(lds_byte), "v"(gaddr)
                 : "memory");
}

__device__ __forceinline__ void wait_asynccnt0() {
#if __has_builtin(__builtin_amdgcn_s_wait_asynccnt)
    __builtin_amdgcn_s_wait_asynccnt(0);
#else
    asm volatile("s_wait_asynccnt 0" ::: "memory");
#endif
}

// ---------------- kernel 2: weighted chunk GEMMs via V_WMMA_F32_16X16X4_F32
__global__ __launch_bounds__(256) void ssd_wmma_kernel(const float* __restrict__ X,
                                                       const float* __restrict__ A,
                                                       const float* __restrict__ Bm,
                                                       const float* __restrict__ Tsuf,
                                                       float* __restrict__ out) {
    int bid = blockIdx.x;
    int cg  = bid % CG;
    int h   = (bid / CG) % NH;
    int b   = bid / (CG * NH);

    int tid  = threadIdx.x;
    int lane = tid & 31;
    int wave = tid >> 5;
    int g    = lane >> 4;     // lane half (K offset +2g in A/B frags)
    int mn   = lane & 15;     // M for A frag / N for B frag

    int tp  = wave & 3;           // p-tile index (0..3)
    int tn0 = (wave >> 2) * 2;    // first of two owned n-tiles

    __shared__ __align__(16) float sX[2][CL * DP];   // double-buffered 16 KB tiles
    __shared__ __align__(16) float sB[2][CL * DN];
    __shared__ float sA[CL];
    __shared__ float sw[CL];

    v8f acc0 = {};
    v8f acc1 = {};

    const size_t rowstride = (size_t)NH * DP;     // floats between consecutive t
    const float* Xb = X  + ((size_t)b * SEQ * NH + h) * DP;
    const float* Bb = Bm + ((size_t)b * SEQ * NH + h) * DN;
    const float* Ab = A  + (size_t)b * SEQ * NH + h;

    // per-thread staging geometry: 4 x b128 per tensor per chunk
    // linear = tid + i*256 -> row = linear>>4 (l), c4 = linear&15 (float4 col)
    unsigned ldsXb[2] = { lds_addr32(&sX[0][0]), lds_addr32(&sX[1][0]) };
    unsigned ldsBb[2] = { lds_addr32(&sB[0][0]), lds_addr32(&sB[1][0]) };

    auto issue_chunk = [&](int buf, int t0) {
        #pragma unroll
        for (int i = 0; i < 4; ++i) {
            int linear = tid + i * 256;      // b128 index 0..1023
            int row = linear >> 4;           // l
            int c4  = linear & 15;           // float4 column
            const float* gx = Xb + (size_t)(t0 + row) * rowstride + c4 * 4;
            const float* gb = Bb + (size_t)(t0 + row) * rowstride + c4 * 4;
            unsigned lo = (unsigned)(linear * 16);
            async_b128_to_lds(ldsXb[buf] + lo, gx);
            async_b128_to_lds(ldsBb[buf] + lo, gb);
        }
    };

    // prologue: async-stage first chunk into buffer 0
    issue_chunk(0, cg * CPB * CL);

    for (int ch = 0; ch < CPB; ++ch) {
        int c   = cg * CPB + ch;
        int t0  = c * CL;
        int buf = ch & 1;

        // stage A chunk (tiny, regular path)
        if (tid < CL) sA[tid] = Ab[(size_t)(t0 + tid) * NH];

        // current buffer's async copies done (this wave), then make visible WG-wide
        wait_asynccnt0();
        __syncthreads();                 // barrier1: buffers ready; prev-iter LDS reads done

        // weights: w[l] = exp(Tsuf_incl[c] - cumsum_{j<=l} A[j])
        if (tid < CL) {
            float cs = 0.0f;
            for (int j = 0; j <= tid; ++j) cs += sA[j];
            float t = Tsuf[((size_t)b * NH + h) * NC + c];
            sw[tid] = __expf(t - cs);
        }

        // overlap: async-stage next chunk into the other buffer during compute
        if (ch + 1 < CPB) issue_chunk(buf ^ 1, t0 + CL);

        __syncthreads();                 // barrier2: sw ready

        // 16 K-steps of 16x16x4 fp32 WMMA; weight folded into A fragment
        #pragma unroll
        for (int ks = 0; ks < CL / 4; ++ks) {
            int k0 = ks * 4 + 2 * g;     // K for frag VGPR0
            int k1 = k0 + 1;             // K for frag VGPR1
            v2f af, bf0, bf1;
            af.x  = sX[buf][k0 * DP + tp * 16 + mn] * sw[k0];
            af.y  = sX[buf][k1 * DP + tp * 16 + mn] * sw[k1];
            bf0.x = sB[buf][k0 * DN + tn0 * 16 + mn];
            bf0.y = sB[buf][k1 * DN + tn0 * 16 + mn];
            bf1.x = sB[buf][k0 * DN + (tn0 + 1) * 16 + mn];
            bf1.y = sB[buf][k1 * DN + (tn0 + 1) * 16 + mn];
            acc0 = __builtin_amdgcn_wmma_f32_16x16x4_f32(false, af, false, bf0,
                                                         (short)0, acc0, false, false);
            acc1 = __builtin_amdgcn_wmma_f32_16x16x4_f32(false, af, false, bf1,
                                                         (short)0, acc1, false, false);
        }
        __syncthreads();                 // protect sA/sw for next iteration
    }

    // writeback: C/D layout VGPR v -> M = v + 8*g, N = mn
    float* ob = out + ((size_t)b * NH + h) * DP * DN;
    #pragma unroll
    for (int v = 0; v < 8; ++v) {
        int p = tp * 16 + v + 8 * g;
        atomicAdd(&ob[p * DN + tn0 * 16 + mn],       acc0[v]);
        atomicAdd(&ob[p * DN + (tn0 + 1) * 16 + mn], acc1[v]);
    }
}

extern "C" void kernel_launch(void* const* d_in, const int* in_sizes, int n_in,
                              void* d_out, int out_size, void* d_ws, size_t ws_size,
                              hipStream_t stream) {
    const float* X  = (const float*)d_in[0];
    const float* A  = (const float*)d_in[1];
    const float* Bm = (const float*)d_in[2];
    float* out  = (float*)d_out;
    float* Tsuf = (float*)d_ws;   // BATCH*NH*NC floats = 64 KB

    zero_out_kernel<<<(out_size + 255) / 256, 256, 0, stream>>>(out, out_size);
    suffix_scan_kernel<<<BATCH * NH, NC, 0, stream>>>(A, Tsuf);
    ssd_wmma_kernel<<<BATCH * NH * CG, 256, 0, stream>>>(X, A, Bm, Tsuf, out);
}